// DurationRegulator_1675037246175
// MI455X (gfx1250) — compile-verified
//
#include <hip/hip_runtime.h>
#include <stdint.h>

// ---------------------------------------------------------------------------
// DurationRegulator for MI455X (gfx1250, wave32)
//   out[b,f,:] = batch[b, upper_bound(csum[b], f), :] * (f < mel_len[b])
//   mel_lens[b] = csum[b, T-1]
// Memory-bound gather: ~100 MB write-once store stream (issued non-temporal,
// th:TH_STORE_NT, to keep the 12.6 MB batch L2-resident for gather reads).
// CDNA5 path: TDM tensor_load_to_lds stages the csum table into LDS.
// ---------------------------------------------------------------------------

#define B_      16
#define T_      512
#define D_      384
#define MAXMEL  4096
#define DV      (D_ / 4)     // 96 float4 per feature row
#define FPB     32           // frames per gather block
#define GTHREADS 512         // 16 waves per gather block
#define BLK_PER_BATCH (MAXMEL / FPB)   // 128

typedef uint32_t u32x4 __attribute__((ext_vector_type(4)));
typedef uint32_t u32x8 __attribute__((ext_vector_type(8)));
typedef float    f32x4 __attribute__((ext_vector_type(4)));

// ---------------------------------------------------------------------------
// Kernel A: masked duration cumsum per batch row (one block per batch).
// ---------------------------------------------------------------------------
__global__ __launch_bounds__(T_) void dr_scan_kernel(
    const int* __restrict__ durs,
    const long long* __restrict__ tok_len,
    const int* __restrict__ alpha,
    int* __restrict__ csum_ws,
    float* __restrict__ mel_out) {
  __shared__ int s[T_];
  const int b = blockIdx.x;
  const int t = threadIdx.x;

  const long long len = tok_len[b];
  int d = durs[b * T_ + t];
  const int a = alpha[0];
  if (a != 1) d = (int)rintf((float)d * (float)a);  // jnp.round = half-to-even
  if ((long long)t >= len) d = 0;
  s[t] = d;
  __syncthreads();

  // Hillis-Steele inclusive scan over 512 elements (9 steps).
  for (int off = 1; off < T_; off <<= 1) {
    const int v = (t >= off) ? s[t - off] : 0;
    __syncthreads();
    s[t] += v;
    __syncthreads();
  }

  csum_ws[b * T_ + t] = s[t];
  if (t == T_ - 1) mel_out[b] = (float)s[t];
}

// ---------------------------------------------------------------------------
// Kernel B: per-frame searchsorted + row gather.
// Wave 0 DMAs the batch's csum (512 x i32 = 2 KB) into LDS via the Tensor
// Data Mover, waits on TENSORcnt, then the block copies 32 frames x 96 float4
// with non-temporal b128 stores.
// ---------------------------------------------------------------------------
__global__ __launch_bounds__(GTHREADS) void dr_gather_kernel(
    const f32x4* __restrict__ batchv,
    const int* __restrict__ csum_ws,
    f32x4* __restrict__ outv) {
  __shared__ int s_csum[T_];
  __shared__ int s_idx[FPB];

  const int b     = blockIdx.x >> 7;            // / BLK_PER_BATCH
  const int fbase = (blockIdx.x & (BLK_PER_BATCH - 1)) * FPB;
  const int tid   = threadIdx.x;

  if (tid < 32) {  // wave 0 only: one TDM descriptor, one DMA per block
    const uint64_t gaddr   = (uint64_t)(uintptr_t)(csum_ws + b * T_);
    const uint32_t lds_off = (uint32_t)(uintptr_t)(&s_csum[0]);

    // D# group 0: count=1 | lds_addr | global_addr[56:0] | type=2
    u32x4 g0;
    g0.x = 1u;                                            // count = 1 descriptor
    g0.y = lds_off;                                       // LDS byte address
    g0.z = (uint32_t)gaddr;                               // global_addr[31:0]
    g0.w = ((uint32_t)(gaddr >> 32) & 0x01FFFFFFu)        // global_addr[56:32]
           | 0x80000000u;                                 // type = 2 ("image")

    // D# group 1: 1-D tile, 512 elems, data_size=4B, no multicast/pad/iterate
    u32x8 g1;
    g1[0] = 0x00020000u;          // data_size = 2 (4 bytes); workgroup_mask = 0
    g1[1] = (uint32_t)T_ << 16;   // tensor_dim0[15:0] at bits 63:48 (=512)
    g1[2] = 1u << 16;             // tensor_dim0[31:16]=0 ; tensor_dim1 lo16 = 1
    g1[3] = (uint32_t)T_ << 16;   // tensor_dim1 hi=0 ; tile_dim0 = 512
    g1[4] = 1u;                   // tile_dim1 = 1 ; tile_dim2 = 0
    g1[5] = (uint32_t)T_;         // tensor_dim0_stride lo32 = 512
    g1[6] = 0u;                   // stride hi / tensor_dim1_stride lo
    g1[7] = 0u;

    asm volatile("tensor_load_to_lds %0, %1" :: "s"(g0), "s"(g1) : "memory");
    __builtin_amdgcn_s_wait_tensorcnt(0);
  }
  __syncthreads();   // csum visible to all 16 waves

  const int mel_len = s_csum[T_ - 1];

  // 32 threads: upper_bound(csum, frame) == searchsorted(..., side='right')
  if (tid < FPB) {
    const int f = fbase + tid;
    int lo = 0, hi = T_;
    while (lo < hi) {
      const int mid = (lo + hi) >> 1;
      if (s_csum[mid] <= f) lo = mid + 1; else hi = mid;
    }
    const int idx = (lo < T_) ? lo : (T_ - 1);            // clip to T-1
    s_idx[tid] = (f < mel_len) ? idx : -1;                // -1 => emit zeros
  }
  __syncthreads();

  // Copy 32 frames x 96 float4 = 3072 b128 transactions with 512 threads.
  // Stores are non-temporal: the output stream is write-once and must not
  // evict the L2-resident batch tensor that the gather loads depend on.
  const int rowbase = b * MAXMEL + fbase;
#pragma unroll
  for (int k = 0; k < (FPB * DV) / GTHREADS; ++k) {       // 6 iterations
    const int l  = k * GTHREADS + tid;
    const int fl = l / DV;
    const int c  = l - fl * DV;
    const int iv = s_idx[fl];
    f32x4 val = {0.f, 0.f, 0.f, 0.f};
    if (iv >= 0) val = batchv[(size_t)(b * T_ + iv) * DV + c];
    __builtin_nontemporal_store(val, &outv[(size_t)(rowbase + fl) * DV + c]);
  }
}

// ---------------------------------------------------------------------------
// Host-side launch (graph-capture safe: only kernel launches on `stream`).
// Inputs: [0] batch f32 (16,512,384)  [1] token_lengths i64 (16,)
//         [2] token_durations i32 (16,512)  [3] alpha scalar
// Output: out f32 (16,4096,384) then mel_lens f32 (16,), concatenated flat.
// ---------------------------------------------------------------------------
extern "C" void kernel_launch(void* const* d_in, const int* in_sizes, int n_in,
                              void* d_out, int out_size, void* d_ws, size_t ws_size,
                              hipStream_t stream) {
  (void)in_sizes; (void)n_in; (void)out_size; (void)ws_size;

  const float*     batch   = (const float*)d_in[0];
  const long long* tok_len = (const long long*)d_in[1];
  const int*       durs    = (const int*)d_in[2];
  const int*       alpha   = (const int*)d_in[3];

  float* out     = (float*)d_out;
  float* mel_out = out + (size_t)B_ * MAXMEL * D_;
  int*   csum_ws = (int*)d_ws;                 // 16*512*4 = 32 KB scratch

  dr_scan_kernel<<<B_, T_, 0, stream>>>(durs, tok_len, alpha, csum_ws, mel_out);
  dr_gather_kernel<<<B_ * BLK_PER_BATCH, GTHREADS, 0, stream>>>(
      (const f32x4*)batch, csum_ws, (f32x4*)out);
}